// HungarianMatcher_74981539053919
// MI455X (gfx1250) — compile-verified
//
#include <hip/hip_runtime.h>
#include <math.h>
#include <stdint.h>

// ---------------------------------------------------------------------------
// DETR HungarianMatcher cost matrix for MI455X (gfx1250).
// Memory-bound (~61 MB moved, ~0.3 GFLOP): no WMMA applies (no contraction).
// CDNA5 path used: async global->LDS DMA (ASYNCcnt) to stage logits rows and
// target boxes/labels, then wave32 shuffle reductions for the softmax stats.
// ---------------------------------------------------------------------------

constexpr int Q    = 300;   // queries
constexpr int T    = 100;   // targets
constexpr int NCLS = 92;    // classes
constexpr int QPB  = 8;     // queries per block (one per wave32)
constexpr int CHNK = (NCLS * 4) / 16;  // 23 x 16B chunks per logits row

// ----- CDNA5 async global->LDS load (B128), inline asm ---------------------
// VGLOBAL encoding: VDST = LDS byte-address VGPR, VADDR = 64-bit global addr.
// Generic LDS pointers keep the LDS offset in the low 32 bits (ISA 10.2).
__device__ __forceinline__ void gld_async_b128(const void* g, void* l) {
  unsigned lds_off = (unsigned)(uintptr_t)l;
  asm volatile("global_load_async_to_lds_b128 %0, %1, off"
               :: "v"(lds_off), "v"(g) : "memory");
}

__device__ __forceinline__ void wait_async0() {
#if __has_builtin(__builtin_amdgcn_s_wait_asynccnt)
  __builtin_amdgcn_s_wait_asynccnt(0);
#else
  asm volatile("s_wait_asynccnt 0" ::: "memory");
#endif
}

// ---------------------------------------------------------------------------

__global__ __launch_bounds__(QPB * 32) void matcher_cost_kernel(
    const float* __restrict__ logits,   // [B, Q, NCLS]
    const float* __restrict__ pboxes,   // [B, Q, 4] cxcywh
    const int*   __restrict__ labels,   // [B, T]
    const float* __restrict__ tboxes,   // [B, T, 4] cxcywh
    float*       __restrict__ out)      // [B, Q, T]
{
  __shared__ __align__(16) float  s_log[QPB][NCLS];  // 8 x 368B (16B aligned rows)
  __shared__ __align__(16) float4 s_tb[T];
  __shared__ __align__(16) int    s_lbl[T];

  const int b    = blockIdx.y;
  const int q0   = blockIdx.x * QPB;
  const int tid  = threadIdx.x;
  const int w    = tid >> 5;
  const int lane = tid & 31;
  const int q    = q0 + w;

  // ---- Stage block-shared data via async DMA (global -> LDS) -------------
  // Target boxes: 100 rows x 16B, one per thread.
  if (tid < T)
    gld_async_b128(tboxes + ((size_t)b * T + tid) * 4, &s_tb[tid]);
  // Target labels: 100 ints = 25 x 16B chunks (threads 128..152).
  if (tid >= 128 && tid < 128 + T / 4)
    gld_async_b128(labels + (size_t)b * T + (size_t)(tid - 128) * 4,
                   &s_lbl[(tid - 128) * 4]);
  // This wave's logits row: 92 floats = 23 x 16B chunks (lanes 0..22).
  if (q < Q && lane < CHNK)
    gld_async_b128(logits + ((size_t)b * Q + q) * NCLS + (size_t)lane * 4,
                   &s_log[w][lane * 4]);

  wait_async0();      // this wave's async transfers have landed in LDS
  __syncthreads();    // make all waves' staged data visible

  if (q >= Q) return;

  // ---- Softmax stats for this query's 92 logits (wave32 butterflies) -----
  const float v0 = s_log[w][lane];
  const float v1 = s_log[w][lane + 32];
  const bool  h2 = (lane + 64) < NCLS;               // lanes 0..27
  const float v2 = h2 ? s_log[w][lane + 64] : -__builtin_inff();

  float m = fmaxf(v0, fmaxf(v1, v2));
#pragma unroll
  for (int o = 16; o; o >>= 1) m = fmaxf(m, __shfl_xor(m, o, 32));

  float e = expf(v0 - m) + expf(v1 - m) + (h2 ? expf(v2 - m) : 0.0f);
#pragma unroll
  for (int o = 16; o; o >>= 1) e += __shfl_xor(e, o, 32);
  const float rinv = 1.0f / e;

  // ---- Pred box (broadcast load) and its xyxy form -----------------------
  const float4 pb = *(const float4*)(pboxes + ((size_t)b * Q + q) * 4);
  const float px0 = pb.x - 0.5f * pb.z, py0 = pb.y - 0.5f * pb.w;
  const float px1 = pb.x + 0.5f * pb.z, py1 = pb.y + 0.5f * pb.w;
  const float parea = (px1 - px0) * (py1 - py0);

  float* __restrict__ orow = out + ((size_t)b * Q + q) * T;

  // ---- Sweep targets; lanes cover t = lane, lane+32, ... (coalesced) -----
  for (int t = lane; t < T; t += 32) {
    // class cost: -softmax(logits)[label]
    const int   lbl  = s_lbl[t];
    const float prob = expf(s_log[w][lbl] - m) * rinv;

    const float4 tb = s_tb[t];
    // L1 on raw cxcywh
    const float cb = fabsf(pb.x - tb.x) + fabsf(pb.y - tb.y) +
                     fabsf(pb.z - tb.z) + fabsf(pb.w - tb.w);

    // GIoU on xyxy
    const float tx0 = tb.x - 0.5f * tb.z, ty0 = tb.y - 0.5f * tb.w;
    const float tx1 = tb.x + 0.5f * tb.z, ty1 = tb.y + 0.5f * tb.w;
    const float tarea = (tx1 - tx0) * (ty1 - ty0);

    const float ix0 = fmaxf(px0, tx0), iy0 = fmaxf(py0, ty0);
    const float ix1 = fminf(px1, tx1), iy1 = fminf(py1, ty1);
    const float iw  = fmaxf(ix1 - ix0, 0.0f), ih = fmaxf(iy1 - iy0, 0.0f);
    const float inter = iw * ih;
    const float uni   = parea + tarea - inter;
    const float iou   = inter / uni;

    const float ex0 = fminf(px0, tx0), ey0 = fminf(py0, ty0);
    const float ex1 = fmaxf(px1, tx1), ey1 = fmaxf(py1, ty1);
    const float ew  = fmaxf(ex1 - ex0, 0.0f), eh = fmaxf(ey1 - ey0, 0.0f);
    const float earea = ew * eh;
    const float giou  = iou - (earea - uni) / earea;

    // C = 5*bbox + 1*(-prob) + 2*(-giou)
    orow[t] = 5.0f * cb - prob - 2.0f * giou;
  }
}

extern "C" void kernel_launch(void* const* d_in, const int* in_sizes, int n_in,
                              void* d_out, int out_size, void* d_ws, size_t ws_size,
                              hipStream_t stream) {
  const float* logits = (const float*)d_in[0];  // [B,Q,92] f32
  const float* pboxes = (const float*)d_in[1];  // [B,Q,4]  f32
  const int*   labels = (const int*)d_in[2];    // [B,T]    i32
  const float* tboxes = (const float*)d_in[3];  // [B,T,4]  f32
  float* out = (float*)d_out;                   // [B,Q,T]  f32

  const int B = in_sizes[0] / (Q * NCLS);       // 256 for the reference setup

  dim3 grid((Q + QPB - 1) / QPB, B);
  dim3 block(QPB * 32);
  matcher_cost_kernel<<<grid, block, 0, stream>>>(logits, pboxes, labels,
                                                  tboxes, out);
}